// MoeReRoutingModel_21741124453165
// MI455X (gfx1250) — compile-verified
//
#include <hip/hip_runtime.h>
#include <hip/hip_bf16.h>
#include <stdint.h>

// MoE re-routing: expert-major/rank-minor permutation of 131072 x 2048 f32 tokens.
// Pure bandwidth problem (~2.15 GB traffic -> ~92us at 23.3 TB/s HBM roofline).
// Main copy uses the CDNA5 async global<->LDS DMA path (ASYNCcnt), with
// non-temporal hints since the stream (2 GB) is far larger than the 192 MB L2.

#define N_TOKENS 131072
#define HIDDEN   2048
#define N_RANKS  8
#define N_EXPERTS 16
#define N_SEG    (N_RANKS * N_EXPERTS)   // 128
#define ROW_BYTES (HIDDEN * 4)           // 8192

// ---------------------------------------------------------------------------
// Kernel A: segment offsets + per-expert totals (trivial serial work, 128 segs)
// ---------------------------------------------------------------------------
__global__ void seg_kernel(const int* __restrict__ counts,   // [R*E] rank-major
                           int* __restrict__ seg_in_off,     // [128] by s = r*E+e
                           int* __restrict__ seg_out_off,    // [128] by s
                           int* __restrict__ seg_cnt,        // [128] by s
                           float* __restrict__ out_expert_num /* [E] */) {
    if (threadIdx.x != 0 || blockIdx.x != 0) return;
    // input offsets: rank-major (given order)
    int acc = 0;
    for (int s = 0; s < N_SEG; ++s) {
        seg_in_off[s] = acc;
        seg_cnt[s]    = counts[s];
        acc += counts[s];
    }
    // output offsets: expert-major, rank-minor (key k = e*R + r)
    acc = 0;
    for (int k = 0; k < N_SEG; ++k) {
        int e = k / N_RANKS;
        int r = k % N_RANKS;
        int s = r * N_EXPERTS + e;
        seg_out_off[s] = acc;
        acc += counts[s];
    }
    // per-expert totals (int32 output emitted value-cast to float)
    for (int e = 0; e < N_EXPERTS; ++e) {
        int sum = 0;
        for (int r = 0; r < N_RANKS; ++r) sum += counts[r * N_EXPERTS + e];
        out_expert_num[e] = (float)sum;
    }
}

// ---------------------------------------------------------------------------
// Kernel B: per-token src-row table, idx output (as float), gathered scales
// ---------------------------------------------------------------------------
__global__ void fill_kernel(const float* __restrict__ scales,
                            const int* __restrict__ seg_in_off,
                            const int* __restrict__ seg_out_off,
                            const int* __restrict__ seg_cnt,
                            int* __restrict__ src_row,        // [N_TOKENS]
                            float* __restrict__ out_scales,   // [N_TOKENS]
                            float* __restrict__ out_idx) {    // [N_TOKENS]
    int k = blockIdx.x;                 // output-order segment
    int e = k / N_RANKS;
    int r = k % N_RANKS;
    int s = r * N_EXPERTS + e;
    int in0  = seg_in_off[s];
    int out0 = seg_out_off[s];
    int n    = seg_cnt[s];
    for (int t = threadIdx.x; t < n; t += blockDim.x) {
        int src = in0 + t;
        int dst = out0 + t;
        src_row[dst]    = src;
        out_idx[dst]    = (float)src;   // exact: src < 2^24
        out_scales[dst] = scales[src];
    }
}

// ---------------------------------------------------------------------------
// Kernel C: 8 KB row gather via async global->LDS->global DMA (ASYNCcnt path)
// One block per output row; 256 threads; each thread moves 2 x 16B chunks.
// Each wave waits only on its own async ops -> no workgroup barrier needed.
// NT hints: stream is read-once/write-once, 10x the size of L2.
// ---------------------------------------------------------------------------
__global__ void copy_kernel(const float* __restrict__ tokens,
                            const int* __restrict__ src_row,
                            float* __restrict__ out_tokens) {
    __shared__ float lds[HIDDEN];       // 8 KB staging

    const int row = blockIdx.x;
    const int src = src_row[row];       // block-uniform
    const float* srcp = tokens + (size_t)src * HIDDEN;
    float* dstp = out_tokens + (size_t)row * HIDDEN;

    const uint32_t tid  = threadIdx.x;
    const uint32_t boff = tid * 16u;                          // byte offset, chunk 0
    // LDS byte address: generic ptr low 32 bits == LDS offset (aperture mapping)
    const uint32_t laddr = (uint32_t)(uintptr_t)(&lds[0]) + boff;

    // async loads: global -> LDS (offset imm applies to BOTH global and LDS addr)
    asm volatile("global_load_async_to_lds_b128 %0, %1, %2 th:TH_LOAD_NT"
                 :: "v"(laddr), "v"(boff), "s"(srcp) : "memory");
    asm volatile("global_load_async_to_lds_b128 %0, %1, %2 offset:4096 th:TH_LOAD_NT"
                 :: "v"(laddr), "v"(boff), "s"(srcp) : "memory");

    asm volatile("s_wait_asynccnt 0" ::: "memory");

    // async stores: LDS -> global
    asm volatile("global_store_async_from_lds_b128 %0, %1, %2 th:TH_STORE_NT"
                 :: "v"(boff), "v"(laddr), "s"(dstp) : "memory");
    asm volatile("global_store_async_from_lds_b128 %0, %1, %2 offset:4096 th:TH_STORE_NT"
                 :: "v"(boff), "v"(laddr), "s"(dstp) : "memory");

    asm volatile("s_wait_asynccnt 0" ::: "memory");
}

// ---------------------------------------------------------------------------
extern "C" void kernel_launch(void* const* d_in, const int* in_sizes, int n_in,
                              void* d_out, int out_size, void* d_ws, size_t ws_size,
                              hipStream_t stream) {
    const float* tokens = (const float*)d_in[0];   // [N_TOKENS*HIDDEN] f32
    const int*   counts = (const int*)d_in[1];     // [R*E] i32
    const float* scales = (const float*)d_in[2];   // [N_TOKENS] f32
    // d_in[3], d_in[4]: scalar mode flags (fixed semantics), unused

    float* out            = (float*)d_out;
    float* out_tokens     = out;                                   // [N_TOKENS*HIDDEN]
    float* out_scales     = out + (size_t)N_TOKENS * HIDDEN;       // [N_TOKENS]
    float* out_idx        = out_scales + N_TOKENS;                 // [N_TOKENS] (value-cast)
    float* out_expert_num = out_idx + N_TOKENS;                    // [N_EXPERTS] (value-cast)

    int* ws          = (int*)d_ws;
    int* seg_in_off  = ws;            // 128
    int* seg_out_off = ws + 128;      // 128
    int* seg_cnt     = ws + 256;      // 128
    int* src_row     = ws + 512;      // N_TOKENS

    seg_kernel<<<1, 32, 0, stream>>>(counts, seg_in_off, seg_out_off, seg_cnt,
                                     out_expert_num);
    fill_kernel<<<N_SEG, 256, 0, stream>>>(scales, seg_in_off, seg_out_off, seg_cnt,
                                           src_row, out_scales, out_idx);
    copy_kernel<<<N_TOKENS, 256, 0, stream>>>(tokens, src_row, out_tokens);
}